// GIN_352187319172
// MI455X (gfx1250) — compile-verified
//
#include <hip/hip_runtime.h>

#define N_NODES 100000
#define N_EDGES 1600000
#define IN_CH 128
#define HID 64
#define OUTD 16
#define NLAYERS 5
#define NGRAPH 512
#define BN_EPS 1e-5f

typedef float v2f __attribute__((ext_vector_type(2)));
typedef float v8f __attribute__((ext_vector_type(8)));
typedef unsigned int v4u __attribute__((ext_vector_type(4)));
typedef int v4i __attribute__((ext_vector_type(4)));
typedef int v8i __attribute__((ext_vector_type(8)));

__device__ __forceinline__ void atomAddF(float* p, float v) {
  // non-returning f32 atomic -> global_atomic_add_f32 (STOREcnt path)
  __hip_atomic_fetch_add(p, v, __ATOMIC_RELAXED, __HIP_MEMORY_SCOPE_AGENT);
}

__global__ void zero_kernel(float* __restrict__ p, int n) {
  int i = blockIdx.x * blockDim.x + threadIdx.x;
  if (i < n) p[i] = 0.0f;
}

// z[i][c] = (1 + eps[layer]) * feat[i][c]
template <int K>
__global__ void init_z_kernel(const float* __restrict__ feat,
                              const float* __restrict__ eps, int layer,
                              float* __restrict__ z) {
  int tid = blockIdx.x * blockDim.x + threadIdx.x;
  const int total = N_NODES * (K / 4);
  if (tid >= total) return;
  float e = 1.0f + eps[layer];
  float4 v = ((const float4*)feat)[tid];
  float4 o = make_float4(v.x * e, v.y * e, v.z * e, v.w * e);
  ((float4*)z)[tid] = o;
}

// z[dst[e]][c] += feat[src[e]][c]  (scatter-add, float4 gather + 4 f32 atomics)
template <int K>
__global__ void edge_agg_kernel(const float* __restrict__ feat,
                                const int* __restrict__ src,
                                const int* __restrict__ dst,
                                float* __restrict__ z) {
  const int chunks = K / 4;
  int tid = blockIdx.x * blockDim.x + threadIdx.x;
  int e = tid / chunks;
  if (e >= N_EDGES) return;
  int c4 = (tid - e * chunks) * 4;
  int s = src[e];
  int d = dst[e];
  float4 v = *(const float4*)(feat + (size_t)s * K + c4);
  float* p = z + (size_t)d * K + c4;
  atomAddF(p + 0, v.x);
  atomAddF(p + 1, v.y);
  atomAddF(p + 2, v.z);
  atomAddF(p + 3, v.w);
}

// Fused: t = relu(BN_a(z @ W1 + b1)); h = relu(BN_b(t @ W2 + b2));
// plus fused graph sum-pool: readout[batch[row]][ro_off + col] += h
// One wave computes a 16x64 output tile using v_wmma_f32_16x16x4_f32.
// The 16xK A-tile is DMA'd into LDS by the Tensor Data Mover with hardware
// row padding (stride K+2 dwords) and hardware zero-fill of tail rows.
template <int K>
__global__ void __launch_bounds__(128)
mlp_fused_kernel(const float* __restrict__ z,
                 const float* __restrict__ W1, const float* __restrict__ b1,
                 const float* __restrict__ ga, const float* __restrict__ bea,
                 const float* __restrict__ ma, const float* __restrict__ va,
                 const float* __restrict__ W2, const float* __restrict__ b2,
                 const float* __restrict__ gb, const float* __restrict__ beb,
                 const float* __restrict__ mb, const float* __restrict__ vb,
                 float* __restrict__ hout, float* __restrict__ readout,
                 const int* __restrict__ batch, int ro_off) {
  constexpr int AS = K + 2;  // LDS row stride for the A tile (bank-friendly)
  __shared__ float lds[4][16 * (IN_CH + 2)];  // 33,280 B total

  const int wave = __builtin_amdgcn_readfirstlane(threadIdx.x >> 5);
  const int lane = threadIdx.x & 31;
  const int half = lane >> 4;   // 0: lanes 0-15, 1: lanes 16-31
  const int l16 = lane & 15;
  const int row0 = blockIdx.x * 64 + wave * 16;
  float* myl = lds[wave];

  // ---- TDM: DMA 16 x K tile of z into LDS (tensor_load_to_lds) ----
  // pad_interval: insert padding every K dwords (K=128 -> code 6, K=64 -> 5)
  // pad_amount  : 2 dwords  -> LDS row stride = K + 2 = AS
  // tensor_dim1 : rows remaining -> hardware zero-fill past end of z
  {
    unsigned lds_off = (unsigned)(size_t)myl;  // low 32 bits = LDS byte offset
    unsigned long long ga64 =
        (unsigned long long)(size_t)(z + (size_t)row0 * K);
    int rows_left = N_NODES - row0;
    if (rows_left < 0) rows_left = 0;
    unsigned td1 = (unsigned)rows_left;
    const unsigned pi = (K == 128) ? 6u : 5u;

    v4u g0;
    g0.x = 1u;                                   // count=1 (valid, user mode)
    g0.y = lds_off;                              // lds_addr
    g0.z = (unsigned)(ga64 & 0xffffffffu);       // global_addr[31:0]
    g0.w = (unsigned)((ga64 >> 32) & 0x01ffffffu) | (2u << 30);  // type=2

    v8i g1;
    g1[0] = (int)((2u << 16) |                   // data_size = 4 bytes
                  (1u << 20) |                   // pad_enable
                  (pi << 22) |                   // pad_interval
                  (1u << 25));                   // pad_amount = 2 dwords
    g1[1] = (int)(((unsigned)K & 0xffffu) << 16);            // tensor_dim0 lo
    g1[2] = (int)(((unsigned)K >> 16) | ((td1 & 0xffffu) << 16));
    g1[3] = (int)((td1 >> 16) | ((unsigned)K << 16));        // tile_dim0 = K
    g1[4] = 16;                                              // tile_dim1 = 16
    g1[5] = K;                                               // dim0_stride lo
    g1[6] = 0;
    g1[7] = 0;

    v4i gz4 = {0, 0, 0, 0};
    v8i gz8 = {0, 0, 0, 0, 0, 0, 0, 0};
    __builtin_amdgcn_tensor_load_to_lds(g0, g1, gz4, gz4, gz8, 0);
    __builtin_amdgcn_s_wait_tensorcnt(0);
  }
  asm volatile("" ::: "memory");  // keep ds_loads below the tensorcnt wait

  // ---- GEMM1: (16 x K) @ (K x 64), K-step 4, fp32 WMMA ----
  v8f acc[4] = {};
  for (int k0 = 0; k0 < K; k0 += 4) {
    v2f a;
    const float* ap = myl + l16 * AS + k0 + half * 2;
    a.x = ap[0];
    a.y = ap[1];
    const float* bp = W1 + (size_t)(k0 + half * 2) * HID + l16;
#pragma unroll
    for (int j = 0; j < 4; ++j) {
      v2f b;
      b.x = bp[j * 16];
      b.y = bp[j * 16 + HID];
      acc[j] = __builtin_amdgcn_wmma_f32_16x16x4_f32(
          false, a, false, b, (short)0, acc[j], false, false);
    }
  }

  // ---- epilogue 1: +b1, BN_a, ReLU; stage 16x64 intermediate in LDS ----
#pragma unroll
  for (int j = 0; j < 4; ++j) {
    int col = j * 16 + l16;
    float bb = b1[col];
    float sc = ga[col] * rsqrtf(va[col] + BN_EPS);
    float sh = bea[col] - ma[col] * sc;
#pragma unroll
    for (int r = 0; r < 8; ++r) {
      float t = (acc[j][r] + bb) * sc + sh;
      t = fmaxf(t, 0.0f);
      myl[(r + half * 8) * 66 + col] = t;  // stride 66: conflict-free-ish
    }
  }

  // ---- GEMM2: (16 x 64) @ (64 x 64) ----
  v8f acc2[4] = {};
  for (int k0 = 0; k0 < HID; k0 += 4) {
    v2f a;
    const float* ap = myl + l16 * 66 + k0 + half * 2;
    a.x = ap[0];
    a.y = ap[1];
    const float* bp = W2 + (size_t)(k0 + half * 2) * HID + l16;
#pragma unroll
    for (int j = 0; j < 4; ++j) {
      v2f b;
      b.x = bp[j * 16];
      b.y = bp[j * 16 + HID];
      acc2[j] = __builtin_amdgcn_wmma_f32_16x16x4_f32(
          false, a, false, b, (short)0, acc2[j], false, false);
    }
  }

  // ---- epilogue 2: +b2, BN_b, ReLU; store h; fused graph pooling ----
  float bb2[4], sc2[4], sh2[4];
#pragma unroll
  for (int j = 0; j < 4; ++j) {
    int col = j * 16 + l16;
    bb2[j] = b2[col];
    sc2[j] = gb[col] * rsqrtf(vb[col] + BN_EPS);
    sh2[j] = beb[col] - mb[col] * sc2[j];
  }
#pragma unroll
  for (int r = 0; r < 8; ++r) {
    int row = row0 + r + half * 8;
    if (row < N_NODES) {
      int g = batch[row];
      float* hrow = hout + (size_t)row * HID;
      float* rrow = readout + (size_t)g * (HID * NLAYERS) + ro_off;
#pragma unroll
      for (int j = 0; j < 4; ++j) {
        int col = j * 16 + l16;
        float t = (acc2[j][r] + bb2[j]) * sc2[j] + sh2[j];
        t = fmaxf(t, 0.0f);
        hrow[col] = t;
        atomAddF(rrow + col, t);
      }
    }
  }
}

// out[g][o] = readout[g][:] @ Wc[:, o] + bc[o]   (G=512, K=320, OUT=16)
__global__ void final_kernel(const float* __restrict__ R,
                             const float* __restrict__ Wc,
                             const float* __restrict__ bc,
                             float* __restrict__ out) {
  int tid = blockIdx.x * blockDim.x + threadIdx.x;
  if (tid >= NGRAPH * OUTD) return;
  int g = tid / OUTD;
  int o = tid - g * OUTD;
  const float* r = R + (size_t)g * (HID * NLAYERS);
  float acc = bc[o];
  for (int k = 0; k < HID * NLAYERS; ++k) acc += r[k] * Wc[k * OUTD + o];
  out[tid] = acc;
}

extern "C" void kernel_launch(void* const* d_in, const int* in_sizes, int n_in,
                              void* d_out, int out_size, void* d_ws,
                              size_t ws_size, hipStream_t stream) {
  const float* x    = (const float*)d_in[0];
  const int*   ei   = (const int*)d_in[1];
  const int*   batch= (const int*)d_in[2];
  const float* eps  = (const float*)d_in[3];
  const float* W1_0 = (const float*)d_in[4];
  const float* b1_0 = (const float*)d_in[5];
  const float* W1_r = (const float*)d_in[6];
  const float* b1_r = (const float*)d_in[7];
  const float* g_a  = (const float*)d_in[8];
  const float* be_a = (const float*)d_in[9];
  const float* m_a  = (const float*)d_in[10];
  const float* v_a  = (const float*)d_in[11];
  const float* W2   = (const float*)d_in[12];
  const float* b2   = (const float*)d_in[13];
  const float* g_b  = (const float*)d_in[14];
  const float* be_b = (const float*)d_in[15];
  const float* m_b  = (const float*)d_in[16];
  const float* v_b  = (const float*)d_in[17];
  const float* Wc   = (const float*)d_in[18];
  const float* bc   = (const float*)d_in[19];
  const int* src = ei;
  const int* dst = ei + N_EDGES;

  float* z       = (float*)d_ws;                       // N x 128
  float* h       = z + (size_t)N_NODES * IN_CH;        // N x 64
  float* readout = h + (size_t)N_NODES * HID;          // G x 320

  zero_kernel<<<(NGRAPH * HID * NLAYERS + 255) / 256, 256, 0, stream>>>(
      readout, NGRAPH * HID * NLAYERS);

  const int mlp_grid = (N_NODES + 63) / 64;

  for (int l = 0; l < NLAYERS; ++l) {
    if (l == 0) {
      int tot = N_NODES * (IN_CH / 4);
      init_z_kernel<IN_CH><<<(tot + 255) / 256, 256, 0, stream>>>(x, eps, 0, z);
      long etot = (long)N_EDGES * (IN_CH / 4);
      edge_agg_kernel<IN_CH><<<(int)((etot + 255) / 256), 256, 0, stream>>>(
          x, src, dst, z);
      mlp_fused_kernel<IN_CH><<<mlp_grid, 128, 0, stream>>>(
          z, W1_0, b1_0, g_a, be_a, m_a, v_a, W2, b2, g_b, be_b, m_b, v_b, h,
          readout, batch, 0);
    } else {
      int tot = N_NODES * (HID / 4);
      init_z_kernel<HID><<<(tot + 255) / 256, 256, 0, stream>>>(h, eps, l, z);
      long etot = (long)N_EDGES * (HID / 4);
      edge_agg_kernel<HID><<<(int)((etot + 255) / 256), 256, 0, stream>>>(
          h, src, dst, z);
      mlp_fused_kernel<HID><<<mlp_grid, 128, 0, stream>>>(
          z, W1_r + (size_t)(l - 1) * HID * HID, b1_r + (size_t)(l - 1) * HID,
          g_a + l * HID, be_a + l * HID, m_a + l * HID, v_a + l * HID,
          W2 + (size_t)l * HID * HID, b2 + l * HID, g_b + l * HID,
          be_b + l * HID, m_b + l * HID, v_b + l * HID, h, readout, batch,
          l * HID);
    }
  }

  final_kernel<<<(NGRAPH * OUTD + 127) / 128, 128, 0, stream>>>(
      readout, Wc, bc, (float*)d_out);
}